// Net_38963943309313
// MI455X (gfx1250) — compile-verified
//
#include <hip/hip_runtime.h>

// ---------------------------------------------------------------------------
// CDNA5 (gfx1250) dense submanifold-style 3D CNN.
//
// Activations are CHANNEL-MAJOR (x[c*V + v]); the implicit GEMM is transposed:
// M = cout, N = voxels, K = 27*cin, using V_WMMA_F32_16X16X4_F32 (full fp32 =>
// numerically matches the f32 reference). Every memory stream is coalesced:
//   A (weights) : repacked fragment order -> one aligned b64 per tile
//   B (acts)    : lanes read consecutive voxels of a channel row -> b32
//   D (stores)  : one channel row x 16 consecutive voxels per reg -> b32
// One wave = 32 couts x 32 voxels (2x2 accumulator tiles): each weight
// fragment feeds two voxel tiles, each activation fragment feeds two cout
// tiles => 4 WMMAs per 6 loads, 4 interleaved accumulate chains.
// All offsets are 32-bit and stepped incrementally (no 64-bit muls in loop);
// all spatial dims are powers of two -> shift/mask indexing, no int div.
// 192MB L2 holds every tensor (max 67MB) => L2-resident; HBM sees only
// compulsory traffic (~0.3GB ~ 15us @ 23.3TB/s) => bound by the f32 WMMA pipe.
// ---------------------------------------------------------------------------

typedef float v2f __attribute__((ext_vector_type(2)));
typedef float v8f __attribute__((ext_vector_type(8)));

#define DD 64  // base grid dimension

// ---------------- scatter: deterministic "highest point index wins" ---------
__global__ void k_zero_u32(unsigned* __restrict__ p, int n) {
  int i = blockIdx.x * blockDim.x + threadIdx.x;
  if (i < n) p[i] = 0u;
}

__global__ void k_scatter_idx(const int* __restrict__ coors,
                              unsigned* __restrict__ idxg, int n) {
  int i = blockIdx.x * blockDim.x + threadIdx.x;
  if (i >= n) return;
  int z = coors[3 * i + 0];
  int y = coors[3 * i + 1];
  int x = coors[3 * i + 2];
  atomicMax(&idxg[(z * DD + y) * DD + x], (unsigned)(i + 1));
}

// Dense input in channel-major layout: f[c*V + v].
__global__ void k_build_dense(const unsigned* __restrict__ idxg,
                              const float* __restrict__ feats,
                              float* __restrict__ f, float* __restrict__ m,
                              int V) {
  int v = blockIdx.x * blockDim.x + threadIdx.x;
  if (v >= V) return;
  unsigned t = idxg[v];
  if (t) {
    int i = (int)t - 1;
    f[0 * V + v] = feats[3 * i + 0];
    f[1 * V + v] = feats[3 * i + 1];
    f[2 * V + v] = feats[3 * i + 2];
    m[v] = 1.0f;
  } else {
    f[0 * V + v] = 0.0f;
    f[1 * V + v] = 0.0f;
    f[2 * V + v] = 0.0f;
    m[v] = 0.0f;
  }
}

// ---------------- weight repack into WMMA A-fragment order ------------------
// Wt index = ((tap*kgroups + kg)*ntiles + nt)*64 + lane*2 + v
// holds W[tap][kg*4 + 2*(lane>>4) + v][nt*16 + (lane&15)], zero-padded in K.
// A tile (16 couts x 4 K): lane (col,g) = row M=col -> co, K = 2g,2g+1.
// => A fragment is one aligned b64 load at (block + lane*8): 256B/wave.
__global__ void k_repack_w(const float* __restrict__ W, float* __restrict__ Wt,
                           int cin, int cout, int total) {
  int idx = blockIdx.x * blockDim.x + threadIdx.x;
  if (idx >= total) return;
  int l2 = idx & 63;
  int v = l2 & 1;
  int lane = l2 >> 1;
  int rest = idx >> 6;
  int ntiles = cout >> 4;
  int nt = rest % ntiles;
  int rest2 = rest / ntiles;
  int kgroups = (cin + 3) >> 2;
  int kg = rest2 % kgroups;
  int tap = rest2 / kgroups;
  int k = kg * 4 + 2 * (lane >> 4) + v;
  int co = nt * 16 + (lane & 15);
  Wt[idx] = (k < cin) ? W[((size_t)tap * cin + k) * cout + co] : 0.0f;
}

// ---------------- implicit-GEMM 3x3x3 SAME conv via f32 WMMA ----------------
// Transposed GEMM: D(cout x vox) = W(cout x K) * X(K x vox) + C.
// Wave32 tile: 32 couts x 32 voxels (acc[ct][vt], ct/vt in {0,1}).
// VGPR layouts (ISA 7.12.2, g = lane>>4, col = lane&15):
//   A (16x4):  row M=col (cout), regs {v0,v1} hold K = 2g, 2g+1
//   B (4x16):  col N=col (voxel), regs {v0,v1} hold K = 2g, 2g+1
//   C (16x16): reg r holds row M = r + 8g (cout), col N = col (voxel)
// RAGGED=true guards the activation loads for cin%4 != 0 (first layer only;
// the weight side is zero-padded by the repack).
template <bool RAGGED>
__global__ __launch_bounds__(32) void k_subm_conv_wmma(
    const float* __restrict__ x, const float* __restrict__ mask,
    const float* __restrict__ Wt, float* __restrict__ y, int ld, int cin,
    int cout) {
  const int D = 1 << ld;
  const int DSQ = D * D;
  const int V = DSQ * D;
  const int lane = threadIdx.x & 31;
  const int col = lane & 15;  // B/C column (voxel), A row (cout) within tile
  const int g = lane >> 4;    // lane group
  const int vox0 = blockIdx.x * 32;
  const int nt0 = blockIdx.y * 2;  // two cout tiles per wave
  const int kgroups = (cin + 3) >> 2;
  const int ntiles = cout >> 4;
  const int wstep = ntiles * 64;  // weight bytes-per-kg step (in floats)

  // This wave's two voxel columns for this lane (shift-only decomposition).
  const int mv0 = vox0 + col;
  const int mv1 = mv0 + 16;
  const bool val0 = (mv0 < V);
  const bool val1 = (mv1 < V);
  const int vz0 = mv0 >> (2 * ld), vy0 = (mv0 >> ld) & (D - 1), vx0 = mv0 & (D - 1);
  const int vz1 = mv1 >> (2 * ld), vy1 = (mv1 >> ld) & (D - 1), vx1 = mv1 & (D - 1);

  v8f acc00 = {0.f, 0.f, 0.f, 0.f, 0.f, 0.f, 0.f, 0.f};  // cout tile0 x vox tile0
  v8f acc01 = acc00;                                      // cout tile0 x vox tile1
  v8f acc10 = acc00;                                      // cout tile1 x vox tile0
  v8f acc11 = acc00;                                      // cout tile1 x vox tile1

  int wtapoff = nt0 * 64 + lane * 2;  // advances by kgroups*wstep per tap
  const int gk = 2 * g;               // lane's K offset within a K=4 group

  for (int dz = -1; dz <= 1; ++dz) {
    const bool zok0 = (unsigned)(vz0 + dz) < (unsigned)D;
    const bool zok1 = (unsigned)(vz1 + dz) < (unsigned)D;
    for (int dy = -1; dy <= 1; ++dy) {
      const bool yok0 = (unsigned)(vy0 + dy) < (unsigned)D;
      const bool yok1 = (unsigned)(vy1 + dy) < (unsigned)D;
      for (int dx = -1; dx <= 1; ++dx) {
        const bool tv0 = val0 && zok0 && yok0 &&
                         (unsigned)(vx0 + dx) < (unsigned)D;
        const bool tv1 = val1 && zok1 && yok1 &&
                         (unsigned)(vx1 + dx) < (unsigned)D;
        const int soff = dz * DSQ + dy * D + dx;  // uniform linear tap shift

        // 32-bit offsets, stepped incrementally (no 64-bit muls in the loop).
        int xoff = mv0 + soff + gk * V;  // x[xoff] = row k=gk, this voxel
        int woff = wtapoff;

#pragma unroll 2
        for (int kg = 0; kg < kgroups; ++kg) {
          v2f b0 = {0.f, 0.f};
          v2f b1 = {0.f, 0.f};
          if (RAGGED) {
            const int k = kg * 4 + gk;
            if (tv0 && k < cin) b0.x = x[xoff];
            if (tv0 && k + 1 < cin) b0.y = x[xoff + V];
            if (tv1 && k < cin) b1.x = x[xoff + 16];
            if (tv1 && k + 1 < cin) b1.y = x[xoff + 16 + V];
          } else {
            if (tv0) {
              b0.x = x[xoff];
              b0.y = x[xoff + V];
            }
            if (tv1) {
              b1.x = x[xoff + 16];
              b1.y = x[xoff + 16 + V];
            }
          }
          // A fragments: one aligned b64 per cout tile (repacked weights).
          const v2f a0 = *(const v2f*)(Wt + woff);
          const v2f a1 = *(const v2f*)(Wt + woff + 64);
          acc00 = __builtin_amdgcn_wmma_f32_16x16x4_f32(
              false, a0, false, b0, (short)0, acc00, false, false);
          acc01 = __builtin_amdgcn_wmma_f32_16x16x4_f32(
              false, a0, false, b1, (short)0, acc01, false, false);
          acc10 = __builtin_amdgcn_wmma_f32_16x16x4_f32(
              false, a1, false, b0, (short)0, acc10, false, false);
          acc11 = __builtin_amdgcn_wmma_f32_16x16x4_f32(
              false, a1, false, b1, (short)0, acc11, false, false);
          xoff += 4 * V;
          woff += wstep;
        }
        wtapoff += kgroups * wstep;
      }
    }
  }

  // Store D with the submanifold mask multiply (y = conv(x) * mask).
  // Reg r = channel nt0*16 + r + 8g (tile0) / +16 (tile1), col = voxel.
  const float mk0 = val0 ? mask[mv0] : 0.0f;
  const float mk1 = val1 ? mask[mv1] : 0.0f;
  int yoff = (nt0 * 16 + 8 * g) * V;  // channel-row base for this lane group
  for (int r = 0; r < 8; ++r) {
    if (val0) {
      y[yoff + mv0] = acc00[r] * mk0;
      y[yoff + 16 * V + mv0] = acc10[r] * mk0;
    }
    if (val1) {
      y[yoff + mv1] = acc01[r] * mk1;
      y[yoff + 16 * V + mv1] = acc11[r] * mk1;
    }
    yoff += V;
  }
}

// ---------------- mask-aware 2x2x2 max pool (channel-major) -----------------
__global__ void k_pool(const float* __restrict__ x, const float* __restrict__ mi,
                       float* __restrict__ y, float* __restrict__ mo, int ld,
                       int C) {
  int idx = blockIdx.x * blockDim.x + threadIdx.x;
  int Do = 1 << ld;
  int Vo = Do * Do * Do;
  if (idx >= Vo * C) return;
  int c = idx >> (3 * ld);   // channel (shift: Vo is a power of two)
  int vox = idx & (Vo - 1);  // output voxel
  int oz = vox >> (2 * ld);
  int oy = (vox >> ld) & (Do - 1);
  int ox = vox & (Do - 1);
  int Di = 2 * Do;
  int Vi = Di * Di * Di;
  float p = -__builtin_inff();
  float mm = 0.0f;
  for (int d = 0; d < 8; ++d) {
    int iz = 2 * oz + (d >> 2);
    int iy = 2 * oy + ((d >> 1) & 1);
    int ix = 2 * ox + (d & 1);
    int iv = (iz * Di + iy) * Di + ix;
    float mk = mi[iv];
    float v = (mk > 0.0f) ? x[c * Vi + iv] : -__builtin_inff();
    p = fmaxf(p, v);
    mm = fmaxf(mm, mk);
  }
  y[c * Vo + vox] = (mm > 0.0f) ? p : 0.0f;
  if (c == 0) mo[vox] = mm;
}

// ---------------------------------------------------------------------------
extern "C" void kernel_launch(void* const* d_in, const int* in_sizes, int n_in,
                              void* d_out, int out_size, void* d_ws,
                              size_t ws_size, hipStream_t stream) {
  const float* features = (const float*)d_in[0];  // (N,3) f32
  const int* coors = (const int*)d_in[1];         // (N,3) i32
  const float* Ws[14];
  for (int i = 0; i < 14; ++i) Ws[i] = (const float*)d_in[2 + i];
  const int N = in_sizes[1] / 3;

  const int V0 = DD * DD * DD;  // 262144

  static const int CIN[14] = {3,   64,  64,  96,  96,  128, 128,
                              160, 160, 192, 192, 224, 224, 256};
  static const int COUT[14] = {64,  64,  96,  96,  128, 128, 160,
                               160, 192, 192, 224, 224, 256, 256};

  // Workspace layout (floats): two ping-pong activation buffers (67MB each),
  // 7 mask levels, u32 scatter-index grid, repacked-weight arena (~41MB).
  float* f0 = (float*)d_ws;
  float* f1 = f0 + (size_t)V0 * 64;
  float* mlev[7];
  float* mp = f1 + (size_t)V0 * 64;
  {
    int d = DD;
    for (int l = 0; l < 7; ++l) {
      mlev[l] = mp;
      mp += d * d * d;
      d >>= 1;
    }
  }
  unsigned* idxg = (unsigned*)mp;
  float* wt_base = (float*)(idxg + V0);
  float* Wt[14];
  int wt_total[14];
  {
    size_t off = 0;
    for (int i = 0; i < 14; ++i) {
      int kgroups = (CIN[i] + 3) >> 2;
      int sz = 27 * kgroups * (COUT[i] >> 4) * 64;
      Wt[i] = wt_base + off;
      wt_total[i] = sz;
      off += sz;
    }
  }

  // 0) weight repack into WMMA fragment order (zero-padded ragged K)
  for (int i = 0; i < 14; ++i) {
    k_repack_w<<<(wt_total[i] + 255) / 256, 256, 0, stream>>>(
        Ws[i], Wt[i], CIN[i], COUT[i], wt_total[i]);
  }

  // 1) deterministic scatter of points into the dense grid
  k_zero_u32<<<(V0 + 255) / 256, 256, 0, stream>>>(idxg, V0);
  k_scatter_idx<<<(N + 255) / 256, 256, 0, stream>>>(coors, idxg, N);
  k_build_dense<<<(V0 + 255) / 256, 256, 0, stream>>>(idxg, features, f0,
                                                      mlev[0], V0);

  // 2) conv / conv / pool pipeline
  float* a = f0;
  float* b = f1;
  int ld = 6;  // log2(D), D = 64
  int lev = 0;
  for (int blk = 0; blk < 7; ++blk) {
    for (int s = 0; s < 2; ++s) {
      const int wi = 2 * blk + s;
      const int ci = CIN[wi], co = COUT[wi];
      const int V = 1 << (3 * ld);
      // Final layer at 1^3 in channel-major is exactly y[c] == d_out layout.
      float* dst = (blk == 6 && s == 1) ? (float*)d_out : b;
      dim3 grid((V + 31) / 32, co / 32);  // wave = 32 couts x 32 voxels
      if (ci & 3) {
        k_subm_conv_wmma<true><<<grid, 32, 0, stream>>>(a, mlev[lev], Wt[wi],
                                                        dst, ld, ci, co);
      } else {
        k_subm_conv_wmma<false><<<grid, 32, 0, stream>>>(a, mlev[lev], Wt[wi],
                                                         dst, ld, ci, co);
      }
      float* t = a;
      a = b;
      b = t;
    }
    if (blk < 6) {
      const int C = COUT[2 * blk + 1];
      const int ldo = ld - 1;
      const int total = (1 << (3 * ldo)) * C;
      k_pool<<<(total + 255) / 256, 256, 0, stream>>>(a, mlev[lev], b,
                                                      mlev[lev + 1], ldo, C);
      float* t = a;
      a = b;
      b = t;
      ld = ldo;
      ++lev;
    }
  }
}